// JastrowUpdate_54984171323908
// MI455X (gfx1250) — compile-verified
//
#include <hip/hip_runtime.h>
#include <math.h>

// CDNA5 (gfx1250) fused kernel:
//   J = occ @ W + b  (via v_wmma_f32_16x16x4_f32, fp32-exact)
//   (sign, logabs) = slogdet(I + J)  (per-lane partial-pivot LU in LDS)
typedef __attribute__((ext_vector_type(2))) float v2f;
typedef __attribute__((ext_vector_type(8))) float v8f;

#define NE   128   // inner dim (n_e)
#define RR   256   // RANK*RANK
#define JPAD 257   // per-matrix LDS stride in floats (conflict-free across lanes)

__global__ __launch_bounds__(32)
void JastrowUpdate_slogdet_kernel(const float* __restrict__ occ,
                                  const float* __restrict__ W,
                                  const float* __restrict__ bias,
                                  float* __restrict__ out,
                                  int Btot) {
    __shared__ float Jlds[32 * JPAD];   // 32 matrices of 16x16, padded

    const int lane = threadIdx.x & 31;
    const int half = lane >> 4;         // 0: lanes 0-15, 1: lanes 16-31
    const int lm   = lane & 15;
    const long row0 = (long)blockIdx.x * 32;
    if (row0 + 32 > Btot) return;       // shapes are multiples of 32

    // ---------------- A panel resident in registers ----------------
    // f32 16x16x4 WMMA A layout (16x4, MxK): lane m = lm, VGPR v holds
    // K = v + 2*half within each K4 block  -> per-lane contiguous float2.
    v2f A[2][32];
#pragma unroll
    for (int t = 0; t < 2; ++t) {
        const float* ap = occ + (row0 + t * 16 + lm) * NE + 2 * half;
#pragma unroll
        for (int kk = 0; kk < 32; ++kk)
            A[t][kk] = *(const v2f*)(ap + 4 * kk);
    }

    // ---------------- GEMM: 16 N-tiles of 16 cols ----------------
    for (int n = 0; n < 16; ++n) {
        v8f acc0 = {};
        v8f acc1 = {};
        // B layout (4x16, KxN): lane col = lm, VGPR v holds K = v + 2*half.
        const float* wp = W + (2 * half) * RR + n * 16 + lm;
#pragma unroll
        for (int kk = 0; kk < 32; ++kk) {
            v2f Bf;
            Bf.x = wp[(4 * kk + 0) * RR];
            Bf.y = wp[(4 * kk + 1) * RR];
            acc0 = __builtin_amdgcn_wmma_f32_16x16x4_f32(
                false, A[0][kk], false, Bf, (short)0, acc0, false, false);
            acc1 = __builtin_amdgcn_wmma_f32_16x16x4_f32(
                false, A[1][kk], false, Bf, (short)0, acc1, false, false);
        }
        // bias + identity, then scatter D tile: acc[v] lane l = element
        // (m = v + 8*half, j = lm) of tile -> matrix q = 16*t + m, row n, col j.
        float bv = bias[n * 16 + lm] + ((lm == n) ? 1.0f : 0.0f);
#pragma unroll
        for (int v = 0; v < 8; ++v) {
            int q0 = v + 8 * half;
            Jlds[q0 * JPAD + n * 16 + lm]        = acc0[v] + bv;
            Jlds[(16 + q0) * JPAD + n * 16 + lm] = acc1[v] + bv;
        }
    }
    __syncthreads();   // single-wave WG: cheap; orders DS before LU reads

    // ---------------- per-lane LU with partial pivoting ----------------
    float* M = &Jlds[lane * JPAD];
    float sign = 1.0f, logabs = 0.0f;
    for (int k = 0; k < 16; ++k) {
        // pivot search (first max, like getrf)
        float best = fabsf(M[k * 16 + k]);
        int p = k;
        for (int i = k + 1; i < 16; ++i) {
            float av = fabsf(M[i * 16 + k]);
            if (av > best) { best = av; p = i; }
        }
        if (p != k) sign = -sign;
        // swap rows k <-> p (cols >= k suffice); p may equal k
        for (int j = k; j < 16; ++j) {
            float a = M[k * 16 + j];
            float c = M[p * 16 + j];
            M[p * 16 + j] = a;
            M[k * 16 + j] = c;
        }
        float piv = M[k * 16 + k];
        if (piv < 0.0f) sign = -sign;
        if (piv == 0.0f) sign = 0.0f;
        logabs += logf(fabsf(piv));
        float inv = 1.0f / piv;
        for (int i = k + 1; i < 16; ++i) {
            float f = M[i * 16 + k] * inv;
            for (int j = k + 1; j < 16; ++j)
                M[i * 16 + j] = fmaf(-f, M[k * 16 + j], M[i * 16 + j]);
        }
    }

    long row = row0 + lane;
    out[row]        = sign;     // j_sign
    out[Btot + row] = logabs;   // j_logabs
}

extern "C" void kernel_launch(void* const* d_in, const int* in_sizes, int n_in,
                              void* d_out, int out_size, void* d_ws, size_t ws_size,
                              hipStream_t stream) {
    const float* occ  = (const float*)d_in[0];   // (B, 128) f32
    const float* W    = (const float*)d_in[1];   // (128, 256) f32
    const float* bias = (const float*)d_in[2];   // (256,) f32
    float* out = (float*)d_out;                  // [sign(B) | logabs(B)]

    int Btot = in_sizes[0] / NE;                 // 262144
    int nblocks = Btot / 32;                     // 8192 waves
    JastrowUpdate_slogdet_kernel<<<dim3(nblocks), dim3(32), 0, stream>>>(
        occ, W, bias, out, Btot);
}